// HandcraftedPodExtractor_50044958933397
// MI455X (gfx1250) — compile-verified
//
#include <hip/hip_runtime.h>

// ---------------------------------------------------------------------------
// HandcraftedPodExtractor for MI455X (gfx1250, wave32)
//
// Pipeline (all bandwidth-bound; input 100.7MB fits in 192MB L2 so pass 2
// re-reads from L2 after pass 1 streams from HBM):
//   K0: zero per-batch bbox accumulators in ws
//   K1: per-batch max|pos| via float4 streaming + LDS tree + u32 atomicMax
//   K2: per-WG 640-float LDS histogram (ds_add_f32), dump partials to ws
//   K3: reduce 16 partials/batch with V_WMMA_F32_16X16X4_F32 (A = ones),
//       then feature scaling + L2 normalization, write (64,640) f32 output
// ---------------------------------------------------------------------------

#define BATCHES        64
#define PTS            65536
#define NBINS          4
#define NCELLS         64
#define NFEAT          10
#define NCOLS          640            // NCELLS * NFEAT
#define WGS_PER_BATCH  16
#define PTS_PER_WG     (PTS / WGS_PER_BATCH)      // 4096
#define K1_BLOCKS      24
#define K1_F4_PER_THR  16
#define FLOATS_PER_B   (PTS * 6)                  // 393216
#define F4_PER_B       (FLOATS_PER_B / 4)         // 98304

typedef float v2f __attribute__((ext_vector_type(2)));
typedef float v8f __attribute__((ext_vector_type(8)));

// ---------------------------------------------------------------- K0: init
__global__ void k0_init(unsigned* __restrict__ bbox) {
  int i = blockIdx.x * blockDim.x + threadIdx.x;
  if (i < BATCHES) bbox[i] = 0u;   // 0u == bits of +0.0f
}

// ------------------------------------------------------- K1: per-batch bbox
// grid = (24, 64), block = 256. Each thread: 16 float4 (strided, coalesced).
__global__ void k1_bbox(const float* __restrict__ x,
                        unsigned* __restrict__ bbox) {
  const int b = blockIdx.y;
  const float4* xb = (const float4*)(x + (size_t)b * FLOATS_PER_B);
  const int base = blockIdx.x * (256 * K1_F4_PER_THR) + threadIdx.x;

  float m = 0.0f;
  for (int i = 0; i < K1_F4_PER_THR; ++i) {
    const int q = base + i * 256;
    __builtin_prefetch(&xb[q + 2048], 0, 0);   // global_prefetch_b8
    const float4 v = xb[q];
    const int m6 = (q * 4) % 6;                // element component phase
    float a0 = (((m6 + 0) % 6) < 3) ? fabsf(v.x) : 0.0f;
    float a1 = (((m6 + 1) % 6) < 3) ? fabsf(v.y) : 0.0f;
    float a2 = (((m6 + 2) % 6) < 3) ? fabsf(v.z) : 0.0f;
    float a3 = (((m6 + 3) % 6) < 3) ? fabsf(v.w) : 0.0f;
    m = fmaxf(m, fmaxf(fmaxf(a0, a1), fmaxf(a2, a3)));
  }

  __shared__ float red[256];
  red[threadIdx.x] = m;
  __syncthreads();
  for (int s = 128; s > 0; s >>= 1) {
    if (threadIdx.x < s) red[threadIdx.x] = fmaxf(red[threadIdx.x], red[threadIdx.x + s]);
    __syncthreads();
  }
  if (threadIdx.x == 0)
    atomicMax(&bbox[b], __float_as_uint(red[0]));  // non-neg floats: bit order == value order
}

// --------------------------------------------- K2: LDS histogram + partials
// grid = (16, 64), block = 256. Each WG: 4096 points -> 640-float LDS hist.
__global__ void k2_hist(const float* __restrict__ x,
                        const unsigned* __restrict__ bbox,
                        float* __restrict__ partial) {
  const int b  = blockIdx.y;
  const int wg = blockIdx.x;

  __shared__ float hist[NCOLS];
  for (int i = threadIdx.x; i < NCOLS; i += 256) hist[i] = 0.0f;
  __syncthreads();

  const float bb     = __uint_as_float(bbox[b]);
  const float inv_th = 1.0f / fmaxf(2.0f * bb, 1e-5f);
  const float* xb = x + (size_t)b * FLOATS_PER_B + (size_t)wg * PTS_PER_WG * 6;

  for (int i = 0; i < PTS_PER_WG / 256; ++i) {     // 16 points / thread
    const float* pp = xb + (size_t)(i * 256 + threadIdx.x) * 6;
    __builtin_prefetch(pp + 256 * 6, 0, 0);
    const float px = pp[0], py = pp[1], pz = pp[2];
    const float ox = pp[3], oy = pp[4], oz = pp[5];
    const float cx = (px + bb) * inv_th;           // pos_in_cube in [0,1]
    const float cy = (py + bb) * inv_th;
    const float cz = (pz + bb) * inv_th;
    const int i0 = min(max((int)(cx * (float)NBINS), 0), NBINS - 1);
    const int i1 = min(max((int)(cy * (float)NBINS), 0), NBINS - 1);
    const int i2 = min(max((int)(cz * (float)NBINS), 0), NBINS - 1);
    const int base = ((i0 * NBINS + i1) * NBINS + i2) * NFEAT;
    // ds_add_f32 (no return) LDS atomics
    atomicAdd(&hist[base + 0], 1.0f);
    atomicAdd(&hist[base + 1], cx);       // sum pos_in_cube; center subtracted in K3
    atomicAdd(&hist[base + 2], cy);
    atomicAdd(&hist[base + 3], cz);
    atomicAdd(&hist[base + 4], ox * ox);
    atomicAdd(&hist[base + 5], ox * oy);
    atomicAdd(&hist[base + 6], ox * oz);
    atomicAdd(&hist[base + 7], oy * oy);
    atomicAdd(&hist[base + 8], oy * oz);
    atomicAdd(&hist[base + 9], oz * oz);
  }
  __syncthreads();

  float* dst = partial + (size_t)(b * WGS_PER_BATCH + wg) * NCOLS;
  for (int i = threadIdx.x; i < NCOLS; i += 256) dst[i] = hist[i];
}

// ----------------------------- K3: WMMA partial reduction + finalize + norm
// grid = 64 (one block per batch), block = 256 (8 waves).
// Each wave reduces 5 column tiles of 16: D = ones(16x4) x partials(4x16) + C,
// chained 4x to fold all 16 partials. With A == ones the result is invariant
// to the B K-row lane layout, so this is layout-robust at full f32 precision.
__global__ void k3_final(const float* __restrict__ partial,
                         float* __restrict__ out) {
  const int b    = blockIdx.x;
  const int lane = threadIdx.x & 31;
  const int wave = threadIdx.x >> 5;

  __shared__ float sums[NCOLS];
  __shared__ float fall[NCOLS];
  __shared__ float red[256];

  const float* pb = partial + (size_t)b * WGS_PER_BATCH * NCOLS;

#if __has_builtin(__builtin_amdgcn_wmma_f32_16x16x4_f32)
  for (int t = wave * 5; t < wave * 5 + 5; ++t) {   // uniform per wave: EXEC all-1s
    const int col = t * 16 + (lane & 15);
    v8f c = {};
    v2f a; a[0] = 1.0f; a[1] = 1.0f;                // all-ones A (16x4)
    for (int k4 = 0; k4 < WGS_PER_BATCH / 4; ++k4) {
      const int r0 = k4 * 4 + ((lane < 16) ? 0 : 2);
      v2f bm;
      bm[0] = pb[(size_t)r0 * NCOLS + col];
      bm[1] = pb[(size_t)(r0 + 1) * NCOLS + col];
      c = __builtin_amdgcn_wmma_f32_16x16x4_f32(false, a, false, bm,
                                                (short)0, c, false, false);
    }
    if (lane < 16) sums[col] = c[0];                // every D row equals the column sum
  }
#else
  for (int t = wave * 5; t < wave * 5 + 5; ++t) {
    const int col = t * 16 + (lane & 15);
    if (lane < 16) {
      float s = 0.0f;
      for (int k = 0; k < WGS_PER_BATCH; ++k) s += pb[(size_t)k * NCOLS + col];
      sums[col] = s;
    }
  }
#endif
  __syncthreads();

  // features: per output i -> cell = i/10, f = i%10
  float ss = 0.0f;
  for (int i = threadIdx.x; i < NCOLS; i += 256) {
    const int cell = i / NFEAT;
    const int f    = i - cell * NFEAT;
    const float cnt  = sums[cell * NFEAT];
    const float fc   = fmaxf(cnt, 1.0f);
    const float upre = rsqrtf(fc);
    float val;
    if (f == 0) {
      val = 0.001f * cnt * upre;
    } else if (f < 4) {
      const int d  = f - 1;
      const int ci = (d == 0) ? (cell >> 4) : ((d == 1) ? ((cell >> 2) & 3) : (cell & 3));
      const float center = ((float)ci + 0.5f) * 0.25f;     // (i+0.5)/NBINS
      val = (sums[i] - cnt * center) * upre;               // sum(offsets) * uprefix
    } else {
      val = sums[i] / fc;                                  // normcov
    }
    fall[i] = val;
    ss += val * val;
  }

  red[threadIdx.x] = ss;
  __syncthreads();
  for (int s = 128; s > 0; s >>= 1) {
    if (threadIdx.x < s) red[threadIdx.x] += red[threadIdx.x + s];
    __syncthreads();
  }
  const float scale = 1.0f / fmaxf(sqrtf(red[0]), 1e-12f);
  for (int i = threadIdx.x; i < NCOLS; i += 256)
    out[(size_t)b * NCOLS + i] = fall[i] * scale;
}

// ---------------------------------------------------------------------------
extern "C" void kernel_launch(void* const* d_in, const int* in_sizes, int n_in,
                              void* d_out, int out_size, void* d_ws, size_t ws_size,
                              hipStream_t stream) {
  const float* x   = (const float*)d_in[0];
  unsigned* bbox   = (unsigned*)d_ws;                 // 64 u32
  float* partial   = (float*)d_ws + 64;               // 64*16*640 f32 = 2.62 MB
  float* out       = (float*)d_out;                   // 64*640 f32

  k0_init <<<1, 64, 0, stream>>>(bbox);
  k1_bbox <<<dim3(K1_BLOCKS, BATCHES), 256, 0, stream>>>(x, bbox);
  k2_hist <<<dim3(WGS_PER_BATCH, BATCHES), 256, 0, stream>>>(x, bbox, partial);
  k3_final<<<BATCHES, 256, 0, stream>>>(partial, out);
}